// VectorQuantizer_vox_68685116998175
// MI455X (gfx1250) — compile-verified
//
#include <hip/hip_runtime.h>
#include <math.h>

typedef float v2f __attribute__((ext_vector_type(2)));
typedef float v8f __attribute__((ext_vector_type(8)));

#define NEMB 512
#define CDIM 128
#define DHW  32768                 // 32*32*32
#define NVEC 262144                // 8 * DHW
#define OUTQ_ELEMS 33554432        // 8*128*DHW
#define LDS_STRIDE 132             // 128 + 4 pad -> conflict-free b64 reads

// ---------------- kernel 0: init (enorm, zero counters) ----------------
__global__ void vq_init(const float* __restrict__ wgt, float* __restrict__ enorm,
                        unsigned* __restrict__ counts, float* __restrict__ sumsq) {
  int e = threadIdx.x;                       // 512 threads
  const float* w = wgt + (size_t)e * CDIM;
  float s = 0.f;
#pragma unroll 8
  for (int c = 0; c < CDIM; ++c) { float v = w[c]; s += v * v; }
  enorm[e]  = s;
  counts[e] = 0u;
  if (e == 0) *sumsq = 0.f;
}

// ---------------- kernel 1: WMMA distance GEMM + argmin ----------------
// One wave = one 16-row tile of n; 8 waves/block share a double-buffered LDS
// stage of each 16-code chunk of the codebook (8x less L2 codebook traffic).
// distance_part = ||e||^2 - 2 * x.e  (||x||^2 is row-constant, dropped)
__global__ __launch_bounds__(256) void vq_argmin(const float* __restrict__ inp,
                                                 const float* __restrict__ wgt,
                                                 const float* __restrict__ enorm,
                                                 int* __restrict__ idxout) {
  __shared__ float smem[2][16 * LDS_STRIDE];

  int t    = threadIdx.x;
  int wave = t >> 5;
  int lane = t & 31;
  int tile = blockIdx.x * 8 + wave;          // 16384 tiles total
  int n0   = tile << 4;
  int b    = n0 >> 15;                       // batch (DHW = 2^15)
  int dhw0 = n0 & (DHW - 1);                 // 16-aligned, never crosses batch
  const float* xb = inp + (size_t)b * ((size_t)CDIM * DHW) + dhw0;

  int l  = lane & 15;                        // row (A) / column e (B,C,D)
  int kh = (lane >> 4) << 1;                 // K pair: 0 for lanes 0-15, 2 for 16-31

  // Cooperative stage geometry: 16 threads/row, 8 channels/thread.
  int sr = t >> 4;                           // 0..15: code row within chunk
  int sc = (t & 15) << 3;                    // channel start

  // Hoist A fragments: 16x128 tile, loaded once, reused for all 32 E-chunks.
  v2f afrag[32];
#pragma unroll
  for (int ks = 0; ks < 32; ++ks) {
    int c0 = ks * 4 + kh;
    afrag[ks].x = xb[(size_t)c0 * DHW + l];
    afrag[ks].y = xb[(size_t)(c0 + 1) * DHW + l];
  }

  float bestd[8];
  int   besti[8];
#pragma unroll
  for (int r = 0; r < 8; ++r) { bestd[r] = 3.4e38f; besti[r] = 0; }

  // Prime buffer 0 with chunk 0 (codes 0..15).
  {
    const float* wrow = wgt + (size_t)sr * CDIM + sc;
    float4 w0 = *(const float4*)(wrow);
    float4 w1 = *(const float4*)(wrow + 4);
    float* dst = &smem[0][sr * LDS_STRIDE + sc];
    *(float4*)dst       = w0;
    *(float4*)(dst + 4) = w1;
  }
  __syncthreads();

  for (int c = 0; c < 32; ++c) {             // 32 chunks of 16 codes
    int cur = c & 1;
    // Stage next chunk into the other buffer while computing this one.
    if (c + 1 < 32) {
      const float* wrow = wgt + (size_t)((c + 1) * 16 + sr) * CDIM + sc;
      float4 w0 = *(const float4*)(wrow);
      float4 w1 = *(const float4*)(wrow + 4);
      float* dst = &smem[cur ^ 1][sr * LDS_STRIDE + sc];
      *(float4*)dst       = w0;
      *(float4*)(dst + 4) = w1;
    }

    const float* wb = &smem[cur][l * LDS_STRIDE + kh];  // this lane's code row
    v8f acc0 = {};
    v8f acc1 = {};
#pragma unroll
    for (int ks = 0; ks < 32; ks += 2) {     // K = 128 in steps of 4, dual chains
      v2f b0 = *(const v2f*)(wb + ks * 4);
      v2f b1 = *(const v2f*)(wb + ks * 4 + 4);
      acc0 = __builtin_amdgcn_wmma_f32_16x16x4_f32(false, afrag[ks],     false, b0,
                                                   (short)0, acc0, false, false);
      acc1 = __builtin_amdgcn_wmma_f32_16x16x4_f32(false, afrag[ks + 1], false, b1,
                                                   (short)0, acc1, false, false);
    }
    int   eidx = c * 16 + l;
    float en   = enorm[eidx];
#pragma unroll
    for (int r = 0; r < 8; ++r) {
      float d = en - 2.0f * (acc0[r] + acc1[r]);
      if (d < bestd[r]) { bestd[r] = d; besti[r] = eidx; }   // strict <: keeps lowest e
    }
    __syncthreads();
  }

  // Reduce min over the 16 columns held by each lane-half (xor masks stay in-half).
#pragma unroll
  for (int r = 0; r < 8; ++r) {
#pragma unroll
    for (int m = 1; m <= 8; m <<= 1) {
      float od = __shfl_xor(bestd[r], m, 32);
      int   oi = __shfl_xor(besti[r], m, 32);
      if (od < bestd[r] || (od == bestd[r] && oi < besti[r])) { bestd[r] = od; besti[r] = oi; }
    }
  }

  // Lane 0 owns rows 0..7, lane 16 owns rows 8..15.
  if ((lane & 15) == 0) {
    int mbase = (lane >> 4) << 3;
#pragma unroll
    for (int r = 0; r < 8; ++r) idxout[n0 + mbase + r] = besti[r];
  }
}

// ---------------- kernel 2: streaming zero-fill of encodings ----------------
// (base only 8B aligned inside the packed output tuple -> v2f, non-temporal)
__global__ void vq_zfill(v2f* __restrict__ p, long long nvec2) {
  long long i      = (long long)blockIdx.x * blockDim.x + threadIdx.x;
  long long stride = (long long)gridDim.x * blockDim.x;
  v2f z = {0.f, 0.f};
  for (; i < nvec2; i += stride) __builtin_nontemporal_store(z, &p[i]);
}

// ---------------- kernel 3: gather, out_q, loss partial, one-hot, histogram ----------------
__global__ __launch_bounds__(256) void vq_gather(const float* __restrict__ inp,
                                                 const float* __restrict__ wgt,
                                                 const int* __restrict__ idx,
                                                 float* __restrict__ outq,
                                                 float* __restrict__ enc,
                                                 unsigned* __restrict__ counts,
                                                 float* __restrict__ sumsq) {
  int n   = blockIdx.x * 256 + threadIdx.x;  // consecutive n across lanes -> coalesced
  int id  = idx[n];
  int b   = n >> 15;
  int dhw = n & (DHW - 1);
  const float* xp = inp  + (size_t)b * ((size_t)CDIM * DHW) + dhw;
  float*       qp = outq + (size_t)b * ((size_t)CDIM * DHW) + dhw;
  const float* w  = wgt  + (size_t)id * CDIM;     // 256 KB codebook: L2-hot
  float s = 0.f;
#pragma unroll 4
  for (int c = 0; c < CDIM; ++c) {
    float wv = w[c];
    float xv = xp[(size_t)c * DHW];
    qp[(size_t)c * DHW] = wv;                // quantized_st forward == quantized
    float d = wv - xv;
    s += d * d;
  }
  atomicAdd(&counts[id], 1u);
  enc[(size_t)n * NEMB + id] = 1.0f;
#pragma unroll
  for (int m = 1; m < 32; m <<= 1) s += __shfl_xor(s, m, 32);
  if ((threadIdx.x & 31) == 0) atomicAdd(sumsq, s);
}

// ---------------- kernel 4: finalize loss + perplexity ----------------
__global__ void vq_final(const unsigned* __restrict__ counts, const float* __restrict__ sumsq,
                         float* __restrict__ loss_out, float* __restrict__ perp_out) {
  __shared__ float red[512];
  int e = threadIdx.x;
  float p = (float)counts[e] * (1.0f / (float)NVEC);
  red[e] = p * logf(p + 1e-10f);
  __syncthreads();
  for (int s = 256; s > 0; s >>= 1) {
    if (e < s) red[e] += red[e + s];
    __syncthreads();
  }
  if (e == 0) {
    *perp_out = expf(-red[0]);
    // loss = q_latent + 0.25*e_latent = 1.25 * mean((q-x)^2) over N*C elements
    *loss_out = 1.25f * (*sumsq) / (float)OUTQ_ELEMS;
  }
}

extern "C" void kernel_launch(void* const* d_in, const int* in_sizes, int n_in,
                              void* d_out, int out_size, void* d_ws, size_t ws_size,
                              hipStream_t stream) {
  const float* inp = (const float*)d_in[0];   // (8,128,32,32,32) f32
  const float* wgt = (const float*)d_in[1];   // (512,128) f32

  float* out  = (float*)d_out;
  float* loss = out;                          // [0]
  float* outq = out + 1;                      // 33554432 elems (8,128,32,32,32)
  float* perp = out + 1 + OUTQ_ELEMS;         // [33554433]
  float* enc  = out + 2 + OUTQ_ELEMS;         // 262144 x 512 one-hot

  float*    enorm  = (float*)d_ws;                       // 512 f32
  unsigned* counts = (unsigned*)((char*)d_ws + 2048);    // 512 u32
  float*    sumsq  = (float*)((char*)d_ws + 4096);       // 1 f32
  int*      idxws  = (int*)((char*)d_ws + 8192);         // 262144 i32

  vq_init  <<<1, 512, 0, stream>>>(wgt, enorm, counts, sumsq);
  vq_argmin<<<2048, 256, 0, stream>>>(inp, wgt, enorm, idxws);
  vq_zfill <<<4096, 256, 0, stream>>>((v2f*)enc, (long long)NVEC * (NEMB / 2));
  vq_gather<<<NVEC / 256, 256, 0, stream>>>(inp, wgt, idxws, outq, enc, counts, sumsq);
  vq_final <<<1, 512, 0, stream>>>(counts, sumsq, loss, perp);
}